// Net_56444460204037
// MI455X (gfx1250) — compile-verified
//
#include <hip/hip_runtime.h>
#include <hip/hip_bf16.h>

// ---------------- Problem constants (match reference) ----------------
#define N_NODES   100000
#define E_EDGES   1200000
#define A_DIM     256
#define H_DIM     64
#define T_TYPES   4
#define STEPS     6
#define B_GRAPHS  64
#define NUM_CLS   2
#define NTILES    (N_NODES / 16)     // 6250, exact
#define NH        ((size_t)N_NODES * H_DIM)

static_assert(N_NODES % 16 == 0, "row tiles must be exact");
static_assert(A_DIM % 4 == 0 && H_DIM % 16 == 0, "K/N tiles exact");

typedef float v2f __attribute__((ext_vector_type(2)));
typedef float v8f __attribute__((ext_vector_type(8)));

// ---------------- WMMA helpers (fp32 16x16x4, exact fp32 math) ----------------
__device__ __forceinline__ v8f wmma4(v2f a, v2f b, v8f c) {
  // 8-arg form: (neg_a, A, neg_b, B, c_mod, C, reuse_a, reuse_b)
  return __builtin_amdgcn_wmma_f32_16x16x4_f32(false, a, false, b, (short)0, c,
                                               false, false);
}

__device__ __forceinline__ v2f load2(const float* p) {
  v2f r; r.x = p[0]; r.y = p[1]; return r;
}

// A fragment: 16x4 tile of row-major X[ld], rows r0..r0+15, cols k0..k0+3.
// lane: M = lane&15, K = k0 + 2*(lane>>4) + v  (contiguous pair -> b64 load)
__device__ __forceinline__ v2f load_afrag(const float* __restrict__ X, int ld,
                                          int r0, int k0, int m, int hf) {
  return load2(X + (size_t)(r0 + m) * ld + k0 + 2 * hf);
}

// B fragment from row-major W[K][Ncols]: B[K=k0+2h+v][N=n0+lane&15]
__device__ __forceinline__ v2f load_bfrag_rm(const float* __restrict__ W, int ld,
                                             int k0, int n0, int m, int hf) {
  v2f b;
  b.x = W[(size_t)(k0 + 2 * hf + 0) * ld + n0 + m];
  b.y = W[(size_t)(k0 + 2 * hf + 1) * ld + n0 + m];
  return b;
}

// B fragment for B = Wt^T where Wt is row-major [J][K] (GRU weights):
// B[K][N=j] = Wt[j][K]  -> contiguous pair within row j.
__device__ __forceinline__ v2f load_bfrag_tr(const float* __restrict__ Wt, int ld,
                                             int k0, int n0, int m, int hf) {
  return load2(Wt + (size_t)(n0 + m) * ld + k0 + 2 * hf);
}

// ---------------- K1: h0 = annotation @ reduce_W + reduce_b ----------------
__global__ void k_reduce(const float* __restrict__ ann, const float* __restrict__ W,
                         const float* __restrict__ bias, float* __restrict__ h0) {
  int wave = (int)((blockIdx.x * blockDim.x + threadIdx.x) >> 5);
  int lane = threadIdx.x & 31;
  if (wave >= NTILES) return;
  int r0 = wave * 16, m = lane & 15, hf = lane >> 4;

  v8f c[4] = {};
  for (int k0 = 0; k0 < A_DIM; k0 += 4) {
    v2f a = load_afrag(ann, A_DIM, r0, k0, m, hf);
#pragma unroll
    for (int j = 0; j < 4; ++j) {
      v2f b = load_bfrag_rm(W, H_DIM, k0, j * 16, m, hf);
      c[j] = wmma4(a, b, c[j]);
    }
  }
#pragma unroll
  for (int j = 0; j < 4; ++j) {
    float bb = bias[j * 16 + m];
#pragma unroll
    for (int v = 0; v < 8; ++v)
      h0[(size_t)(r0 + v + 8 * hf) * H_DIM + j * 16 + m] = c[j][v] + bb;
  }
}

// -------- K2: tr[t] = h @ edge_W[t] + edge_b[t], one wave per (tile,type) -----
__global__ void k_transform(const float* __restrict__ h, const float* __restrict__ eW,
                            const float* __restrict__ eb, float* __restrict__ tr) {
  int task = (int)((blockIdx.x * blockDim.x + threadIdx.x) >> 5);
  int lane = threadIdx.x & 31;
  if (task >= NTILES * T_TYPES) return;
  int t = task & 3, tile = task >> 2;
  int r0 = tile * 16, m = lane & 15, hf = lane >> 4;
  const float* W = eW + (size_t)t * H_DIM * H_DIM;

  v8f c[4] = {};
  for (int k0 = 0; k0 < H_DIM; k0 += 4) {
    v2f a = load_afrag(h, H_DIM, r0, k0, m, hf);
#pragma unroll
    for (int j = 0; j < 4; ++j) {
      v2f b = load_bfrag_rm(W, H_DIM, k0, j * 16, m, hf);
      c[j] = wmma4(a, b, c[j]);
    }
  }
  float* out = tr + (size_t)t * N_NODES * H_DIM;
#pragma unroll
  for (int j = 0; j < 4; ++j) {
    float bb = eb[t * H_DIM + j * 16 + m];
#pragma unroll
    for (int v = 0; v < 8; ++v)
      out[(size_t)(r0 + v + 8 * hf) * H_DIM + j * 16 + m] = c[j][v] + bb;
  }
}

// ---------------- K3: zero fill ----------------
__global__ void k_zero(float* __restrict__ p, long n) {
  long i = (long)blockIdx.x * blockDim.x + threadIdx.x;
  if (i < n) p[i] = 0.0f;
}

// ------- K4: per-edge gather + scatter-add (wave per edge, 2 cols/lane) -------
__global__ void k_scatter(const float* __restrict__ tr, const int* __restrict__ src,
                          const int* __restrict__ dst, const int* __restrict__ et,
                          float* __restrict__ agg) {
  int e = (int)((blockIdx.x * blockDim.x + threadIdx.x) >> 5);
  int lane = threadIdx.x & 31;
  if (e >= E_EDGES) return;
  int s = src[e], d = dst[e], t = et[e];
  const float* row = tr + ((size_t)t * N_NODES + s) * H_DIM;
  v2f v = load2(row + 2 * lane);
  float* o = agg + (size_t)d * H_DIM + 2 * lane;
  atomicAdd(o + 0, v.x);
  atomicAdd(o + 1, v.y);
}

// ---------------- K5: fused GRU cell (wave per 16-row tile) ----------------
__global__ void k_gru(const float* __restrict__ agg, const float* __restrict__ hin,
                      const float* __restrict__ Wi, const float* __restrict__ bi,
                      const float* __restrict__ Wh, const float* __restrict__ bh,
                      float* __restrict__ hout) {
  int wave = (int)((blockIdx.x * blockDim.x + threadIdx.x) >> 5);
  int lane = threadIdx.x & 31;
  if (wave >= NTILES) return;
  int r0 = wave * 16, m = lane & 15, hf = lane >> 4;

  for (int jc = 0; jc < 4; ++jc) {       // 16-col tile of the H=64 output
    v8f gi[3] = {};                      // r, z, n input-gates  (agg @ Wi^T)
    v8f gh[3] = {};                      // r, z, n hidden-gates (h   @ Wh^T)
    for (int k0 = 0; k0 < H_DIM; k0 += 4) {
      v2f aa = load_afrag(agg, H_DIM, r0, k0, m, hf);
      v2f ah = load_afrag(hin, H_DIM, r0, k0, m, hf);
#pragma unroll
      for (int g = 0; g < 3; ++g) {
        int n0 = g * H_DIM + jc * 16;    // col block in [0,192)
        v2f bwi = load_bfrag_tr(Wi, H_DIM, k0, n0, m, hf);
        gi[g] = wmma4(aa, bwi, gi[g]);
        v2f bwh = load_bfrag_tr(Wh, H_DIM, k0, n0, m, hf);
        gh[g] = wmma4(ah, bwh, gh[g]);
      }
    }
    int col = jc * 16 + m;
    float biR = bi[0 * H_DIM + col], biZ = bi[1 * H_DIM + col], biN = bi[2 * H_DIM + col];
    float bhR = bh[0 * H_DIM + col], bhZ = bh[1 * H_DIM + col], bhN = bh[2 * H_DIM + col];
#pragma unroll
    for (int v = 0; v < 8; ++v) {
      int row = r0 + v + 8 * hf;
      float ir = gi[0][v] + biR, iz = gi[1][v] + biZ, in_ = gi[2][v] + biN;
      float hr = gh[0][v] + bhR, hz = gh[1][v] + bhZ, hn = gh[2][v] + bhN;
      float r = 1.0f / (1.0f + expf(-(ir + hr)));
      float z = 1.0f / (1.0f + expf(-(iz + hz)));
      float nn = tanhf(in_ + r * hn);
      float hp = hin[(size_t)row * H_DIM + col];
      hout[(size_t)row * H_DIM + col] = (1.0f - z) * nn + z * hp;
    }
  }
}

// ---------------- pooling: monotone uint key for float atomicMax ----------------
__device__ __forceinline__ unsigned fkey(float x) {
  unsigned u = __float_as_uint(x);
  return (u & 0x80000000u) ? ~u : (u | 0x80000000u);
}

// K6: gate score per node (wave per node) + segment atomicMax
__global__ void k_gate(const float* __restrict__ h, const float* __restrict__ h0,
                       const int* __restrict__ gid, const float* __restrict__ gW,
                       const float* __restrict__ gb, float* __restrict__ gate,
                       unsigned* __restrict__ gmaxk) {
  int n = (int)((blockIdx.x * blockDim.x + threadIdx.x) >> 5);
  int lane = threadIdx.x & 31;
  if (n >= N_NODES) return;
  const float* hr = h + (size_t)n * H_DIM;
  const float* h0r = h0 + (size_t)n * H_DIM;
  float s = hr[lane] * gW[lane] + hr[lane + 32] * gW[lane + 32] +
            h0r[lane] * gW[64 + lane] + h0r[lane + 32] * gW[96 + lane];
#pragma unroll
  for (int off = 16; off; off >>= 1) s += __shfl_xor(s, off, 32);
  if (lane == 0) {
    float g = s + gb[0];
    gate[n] = g;
    atomicMax(&gmaxk[gid[n]], fkey(g));
  }
}

// K7: decode per-graph max keys, guard non-finite -> 0 (matches reference)
__global__ void k_gmax(const unsigned* __restrict__ gmaxk, float* __restrict__ gmax) {
  int b = threadIdx.x;
  if (b >= B_GRAPHS) return;
  unsigned k = gmaxk[b];
  unsigned u = (k >> 31) ? (k ^ 0x80000000u) : ~k;
  float x = __uint_as_float(u);
  gmax[b] = __builtin_isfinite(x) ? x : 0.0f;
}

// K8: e = exp(gate-gmax); denom += e; readout += e*[h,h0]   (wave per node)
__global__ void k_pool(const float* __restrict__ h, const float* __restrict__ h0,
                       const int* __restrict__ gid, const float* __restrict__ gate,
                       const float* __restrict__ gmax, float* __restrict__ denom,
                       float* __restrict__ readout) {
  int n = (int)((blockIdx.x * blockDim.x + threadIdx.x) >> 5);
  int lane = threadIdx.x & 31;
  if (n >= N_NODES) return;
  int g = gid[n];
  float ev = expf(gate[n] - gmax[g]);
  if (lane == 0) atomicAdd(&denom[g], ev);
  const float* hr = h + (size_t)n * H_DIM;
  const float* h0r = h0 + (size_t)n * H_DIM;
  float* ro = readout + (size_t)g * (2 * H_DIM);
  atomicAdd(&ro[lane],       ev * hr[lane]);
  atomicAdd(&ro[lane + 32],  ev * hr[lane + 32]);
  atomicAdd(&ro[64 + lane],  ev * h0r[lane]);
  atomicAdd(&ro[96 + lane],  ev * h0r[lane + 32]);
}

// K9: logits[b][c] = (readout[b]/denom[b]) @ out_W + out_b
__global__ void k_logits(const float* __restrict__ readout, const float* __restrict__ denom,
                         const float* __restrict__ oW, const float* __restrict__ ob,
                         float* __restrict__ out) {
  int t = threadIdx.x;
  if (t >= B_GRAPHS * NUM_CLS) return;
  int b = t >> 1, c = t & 1;
  float d = denom[b];
  float scale = (d > 0.0f) ? (1.0f / d) : 0.0f;
  float acc = ob[c];
  for (int k = 0; k < 2 * H_DIM; ++k)
    acc += readout[b * (2 * H_DIM) + k] * scale * oW[k * NUM_CLS + c];
  out[t] = acc;
}

// ---------------- host: launch the pipeline ----------------
extern "C" void kernel_launch(void* const* d_in, const int* in_sizes, int n_in,
                              void* d_out, int out_size, void* d_ws, size_t ws_size,
                              hipStream_t stream) {
  (void)in_sizes; (void)n_in; (void)out_size; (void)ws_size;
  const float* ann   = (const float*)d_in[0];
  const int*   src   = (const int*)d_in[1];
  const int*   dst   = (const int*)d_in[2];
  const int*   etyp  = (const int*)d_in[3];
  const int*   gid   = (const int*)d_in[4];
  const float* rW    = (const float*)d_in[5];
  const float* rb    = (const float*)d_in[6];
  const float* eW    = (const float*)d_in[7];
  const float* eb    = (const float*)d_in[8];
  const float* gWi   = (const float*)d_in[9];
  const float* gbi   = (const float*)d_in[10];
  const float* gWh   = (const float*)d_in[11];
  const float* gbh   = (const float*)d_in[12];
  const float* gateW = (const float*)d_in[13];
  const float* gateB = (const float*)d_in[14];
  const float* outW  = (const float*)d_in[15];
  const float* outB  = (const float*)d_in[16];
  float* out = (float*)d_out;

  // workspace layout (floats)
  float* ws   = (float*)d_ws;
  float* h0   = ws;                 // N*H
  float* hA   = h0 + NH;            // N*H
  float* hB   = hA + NH;            // N*H
  float* agg  = hB + NH;            // N*H
  float* tr   = agg + NH;           // T*N*H
  float* gate = tr + (size_t)T_TYPES * NH;              // N
  unsigned* gmaxk = (unsigned*)(gate + N_NODES);        // B
  float* gmax    = (float*)(gmaxk + B_GRAPHS);          // B
  float* denom   = gmax + B_GRAPHS;                     // B
  float* readout = denom + B_GRAPHS;                    // B*2H

  const int TB = 256;
  // init pooling accumulators (gmaxk as 0u == 0.0f bits; denom+readout zeros)
  k_zero<<<1, TB, 0, stream>>>((float*)gmaxk, B_GRAPHS);
  long zr = B_GRAPHS + (long)B_GRAPHS * 2 * H_DIM;
  k_zero<<<(int)((zr + TB - 1) / TB), TB, 0, stream>>>(denom, zr);

  // h0 = ann @ rW + rb
  k_reduce<<<(NTILES * 32 + TB - 1) / TB, TB, 0, stream>>>(ann, rW, rb, h0);

  // 6 propagation steps, ping-pong h buffers
  const float* hcur = h0;
  float* bufs[2] = {hA, hB};
  for (int s = 0; s < STEPS; ++s) {
    float* hnext = bufs[s & 1];
    k_transform<<<(NTILES * T_TYPES * 32 + TB - 1) / TB, TB, 0, stream>>>(hcur, eW, eb, tr);
    k_zero<<<(int)((NH + TB - 1) / TB), TB, 0, stream>>>(agg, (long)NH);
    k_scatter<<<(E_EDGES * 32 + TB - 1) / TB, TB, 0, stream>>>(tr, src, dst, etyp, agg);
    k_gru<<<(NTILES * 32 + TB - 1) / TB, TB, 0, stream>>>(agg, hcur, gWi, gbi, gWh, gbh, hnext);
    hcur = hnext;
  }

  // attention pooling + logits
  k_gate<<<(N_NODES * 32 + TB - 1) / TB, TB, 0, stream>>>(hcur, h0, gid, gateW, gateB, gate, gmaxk);
  k_gmax<<<1, B_GRAPHS, 0, stream>>>(gmaxk, gmax);
  k_pool<<<(N_NODES * 32 + TB - 1) / TB, TB, 0, stream>>>(hcur, h0, gid, gate, gmax, denom, readout);
  k_logits<<<1, 128, 0, stream>>>(readout, denom, outW, outB, out);
}